// CandidateAwareAggregation_17377437680000
// MI455X (gfx1250) — compile-verified
//
#include <hip/hip_runtime.h>

typedef __attribute__((ext_vector_type(2))) float v2f;
typedef __attribute__((ext_vector_type(4))) float v4f;
typedef __attribute__((ext_vector_type(8))) float v8f;

#define BSZ 1024
#define KK  32
#define CC  50
#define DD  128
#define LDSTR 132   // LDS row pitch (floats): 528B, %16==0, breaks bank periodicity

__device__ __forceinline__ float fast_tanh(float x) {
#if __has_builtin(__builtin_amdgcn_tanhf)
  return __builtin_amdgcn_tanhf(x);          // v_tanh_f32 (gfx1250 trans op)
#else
  float e = __expf(2.0f * x);                // v_exp_f32
  return 1.0f - 2.0f * __builtin_amdgcn_rcpf(e + 1.0f);  // v_rcp_f32
#endif
}

// Async 16B global->LDS copy, tracked by ASYNCcnt (no VGPR round-trip).
__device__ __forceinline__ void async_copy_b128(unsigned lds_off, const float* gptr) {
  asm volatile("global_load_async_to_lds_b128 %0, %1, off"
               :: "v"(lds_off), "v"(gptr) : "memory");
}
__device__ __forceinline__ void wait_async0() {
  asm volatile("s_wait_asynccnt 0x0" ::: "memory");
}

// ---------------------------------------------------------------------------
// Kernel A: pi = IV @ Wi (+b1 fused), pc = CV @ Wc, one tiled f32 WMMA GEMM
// over the concatenated row space. 8 waves/block, each owns a 16x16 N-tile,
// K-loop = 32 x V_WMMA_F32_16X16X4_F32. Staging via async-to-LDS B128.
// ---------------------------------------------------------------------------
__global__ __launch_bounds__(256) void proj_gemm(
    const float* __restrict__ iv, const float* __restrict__ cv,
    const float* __restrict__ W1, const float* __restrict__ b1,
    float* __restrict__ pi, float* __restrict__ pc) {
  extern __shared__ float smem[];
  float* As = smem;                 // 16 x LDSTR  (A tile, row major)
  float* Ws = smem + 16 * LDSTR;    // 128 x LDSTR (W half, row major: Ws[d][e])

  const int PI_TILES = (BSZ * KK) / 16;   // 2048
  int tile = blockIdx.x;
  bool isPi = tile < PI_TILES;
  const float* src = isPi ? iv : cv;
  float* dst       = isPi ? pi : pc;
  int row0 = (isPi ? tile : tile - PI_TILES) * 16;
  const float* Wsrc = W1 + (isPi ? 0 : DD * DD);   // Wi = W1[:128], Wc = W1[128:]

  int tid = threadIdx.x;
  unsigned as_off = (unsigned)(unsigned long long)As;
  unsigned ws_off = (unsigned)(unsigned long long)Ws;

  // A tile: 16 rows x 32 float4 (async, 2 per thread)
  for (int idx = tid; idx < 16 * 32; idx += 256) {
    int r = idx >> 5, c4 = (idx & 31) * 4;
    async_copy_b128(as_off + (unsigned)(r * LDSTR + c4) * 4u,
                    src + (size_t)(row0 + r) * DD + c4);
  }
  // W half: 128 rows x 32 float4 (async, 16 per thread)
  for (int idx = tid; idx < DD * 32; idx += 256) {
    int d = idx >> 5, c4 = (idx & 31) * 4;
    async_copy_b128(ws_off + (unsigned)(d * LDSTR + c4) * 4u,
                    Wsrc + (size_t)d * DD + c4);
  }
  wait_async0();        // own-wave async complete before barrier
  __syncthreads();

  int wave = tid >> 5, lane = tid & 31;
  int n15  = lane & 15;
  int koff = (lane >> 4) * 2;       // ISA A/B frag layout: lanes16-31 hold K=2,3
  int M = n15;                      // A row within tile
  int N = wave * 16 + n15;          // output column

  v8f acc = {};
#pragma unroll 4
  for (int k4 = 0; k4 < DD; k4 += 4) {
    v2f a = *(const v2f*)&As[M * LDSTR + k4 + koff];        // ds_load_b64
    v2f b;                                                   // two b32, 132-dword pitch
    b[0] = Ws[(k4 + koff) * LDSTR + N];
    b[1] = Ws[(k4 + koff + 1) * LDSTR + N];
    acc = __builtin_amdgcn_wmma_f32_16x16x4_f32(
        false, a, false, b, (short)0, acc, false, false);
  }
  // D layout: VGPR r -> M = r (lanes 0-15) / r+8 (lanes 16-31)
  float bias = isPi ? b1[N] : 0.0f;     // fuse hidden-layer bias into pi
  int mbase = (lane < 16) ? 0 : 8;
#pragma unroll
  for (int r = 0; r < 8; ++r)
    dst[(size_t)(row0 + mbase + r) * DD + N] = acc[r] + bias;
}

// ---------------------------------------------------------------------------
// Kernel B: one block per batch. Async-staged LDS tiles; tanh-attention +
// softmax + WMMA dot_scores (iv x cv^T) + weighted aggregation.
// ---------------------------------------------------------------------------
__global__ __launch_bounds__(256) void fused_attn(
    const float* __restrict__ iv_g, const float* __restrict__ cv_g,
    const float* __restrict__ W2, const float* __restrict__ b2,
    const float* __restrict__ pi_ws, const float* __restrict__ pc_ws,
    float* __restrict__ out) {
  extern __shared__ float smem[];
  float* piB = smem;                    // 32 x LDSTR   (pi + b1, pre-fused)
  float* pcL = piB + KK * LDSTR;        // 50 x LDSTR
  float* ivL = pcL + CC * LDSTR;        // 32 x LDSTR
  float* cvL = ivL + KK * LDSTR;        // 64 x LDSTR   (rows 50..63 zeroed)
  float* w2L = cvL + 64 * LDSTR;        // 128
  float* sc  = w2L + DD;                // 32 x 52
  float* wgt = sc + KK * 52;            // 32 x 52
  float* acc = wgt + KK * 52;           // 64

  int b = blockIdx.x, tid = threadIdx.x;
  unsigned piB_o = (unsigned)(unsigned long long)piB;
  unsigned pcL_o = (unsigned)(unsigned long long)pcL;
  unsigned ivL_o = (unsigned)(unsigned long long)ivL;
  unsigned cvL_o = (unsigned)(unsigned long long)cvL;

  // Async stage: pure B128 copies (pi already has b1 folded in by kernel A)
  for (int idx = tid; idx < KK * 32; idx += 256) {      // 1024 float4 each
    int r = idx >> 5, c4 = (idx & 31) * 4;
    unsigned lo = (unsigned)(r * LDSTR + c4) * 4u;
    async_copy_b128(piB_o + lo, pi_ws + ((size_t)b * KK + r) * DD + c4);
    async_copy_b128(ivL_o + lo, iv_g + ((size_t)b * KK + r) * DD + c4);
  }
  for (int idx = tid; idx < CC * 32; idx += 256) {      // 1600 float4 each
    int r = idx >> 5, c4 = (idx & 31) * 4;
    unsigned lo = (unsigned)(r * LDSTR + c4) * 4u;
    async_copy_b128(pcL_o + lo, pc_ws + ((size_t)b * CC + r) * DD + c4);
    async_copy_b128(cvL_o + lo, cv_g + ((size_t)b * CC + r) * DD + c4);
  }
  for (int idx = tid; idx < (64 - CC) * DD; idx += 256) {
    int c = CC + (idx >> 7), d = idx & 127;
    cvL[c * LDSTR + d] = 0.0f;          // zero pad for WMMA N-tile 3
  }
  if (tid < DD) w2L[tid] = W2[tid];
  if (tid < 64) acc[tid] = 0.0f;
  wait_async0();        // own-wave async complete before barrier
  __syncthreads();

  float b2v = b2[0];
  int wave = tid >> 5, lane = tid & 31;

  // ---- Stage 2: attn_scores[k][c] = b2 + sum_d W2[d]*tanh(piB[k][d]+pc[c][d])
  for (int kk = 0; kk < 4; ++kk) {
    int k = wave + kk * 8;
    for (int cpass = 0; cpass < 2; ++cpass) {
      int c = lane + cpass * 32;
      if (c < CC) {
        float s = b2v;
#pragma unroll 4
        for (int d = 0; d < DD; d += 4) {
          v4f p = *(const v4f*)&piB[k * LDSTR + d];   // wave-uniform broadcast
          v4f q = *(const v4f*)&pcL[c * LDSTR + d];   // per-lane row, padded pitch
          v4f w = *(const v4f*)&w2L[d];
          s += w[0] * fast_tanh(p[0] + q[0]);
          s += w[1] * fast_tanh(p[1] + q[1]);
          s += w[2] * fast_tanh(p[2] + q[2]);
          s += w[3] * fast_tanh(p[3] + q[3]);
        }
        sc[k * 52 + c] = s;
      }
    }
  }
  __syncthreads();

  // ---- Stage 3: softmax over k (32 values) per candidate c
  if (tid < CC) {
    int c = tid;
    float m = -3.0e38f;
#pragma unroll
    for (int k = 0; k < KK; ++k) m = fmaxf(m, sc[k * 52 + c]);
    float ex[KK], sum = 0.0f;
#pragma unroll
    for (int k = 0; k < KK; ++k) { float e = __expf(sc[k * 52 + c] - m); ex[k] = e; sum += e; }
    float inv = __builtin_amdgcn_rcpf(sum);
#pragma unroll
    for (int k = 0; k < KK; ++k) wgt[k * 52 + c] = ex[k] * inv;
  }
  __syncthreads();

  // ---- Stage 4: dot[k][c] = iv[k] . cv[c] via WMMA f32; 8 waves = 2x4 tiles
  int mi = wave >> 2, ni = wave & 3;
  int n15 = lane & 15, koff = (lane >> 4) * 2;
  int M = mi * 16 + n15;
  int N = ni * 16 + n15;
  v8f dfrag = {};
#pragma unroll 4
  for (int k4 = 0; k4 < DD; k4 += 4) {
    v2f a = *(const v2f*)&ivL[M * LDSTR + k4 + koff];
    v2f bb = *(const v2f*)&cvL[N * LDSTR + k4 + koff];
    dfrag = __builtin_amdgcn_wmma_f32_16x16x4_f32(
        false, a, false, bb, (short)0, dfrag, false, false);
  }
  // weighted reduction over k: each lane holds 8 rows of its column N
  int mbase = mi * 16 + ((lane < 16) ? 0 : 8);
  int Ncl = (N < CC) ? N : (CC - 1);
  float nval = (N < CC) ? 1.0f : 0.0f;     // arithmetic mask; EXEC stays full
  float psum = 0.0f;
#pragma unroll
  for (int r = 0; r < 8; ++r)
    psum += (wgt[(mbase + r) * 52 + Ncl] * nval) * dfrag[r];
  psum += __shfl_xor(psum, 16, 32);        // fold lanes l <-> l+16 (rows M vs M+8)
  if (lane < 16) atomicAdd(&acc[ni * 16 + n15], psum);   // ds_add_f32
  __syncthreads();

  if (tid < CC) out[(size_t)b * CC + tid] = acc[tid];
}

extern "C" void kernel_launch(void* const* d_in, const int* in_sizes, int n_in,
                              void* d_out, int out_size, void* d_ws, size_t ws_size,
                              hipStream_t stream) {
  const float* iv = (const float*)d_in[0];
  const float* cv = (const float*)d_in[1];
  const float* W1 = (const float*)d_in[2];
  const float* b1 = (const float*)d_in[3];
  const float* W2 = (const float*)d_in[4];
  const float* b2 = (const float*)d_in[5];
  float* out = (float*)d_out;

  float* pi = (float*)d_ws;                          // 1024*32*128 f32
  float* pc = pi + (size_t)BSZ * KK * DD;            // 1024*50*128 f32

  size_t smemA = (size_t)(16 + DD) * LDSTR * sizeof(float);           // ~76 KB
  size_t smemB = (size_t)(KK * LDSTR + CC * LDSTR + KK * LDSTR +
                          64 * LDSTR + DD + KK * 52 + KK * 52 + 64) *
                 sizeof(float);                                       // ~108 KB

  int tilesA = (BSZ * KK + BSZ * CC) / 16;           // 5248
  proj_gemm<<<tilesA, 256, smemA, stream>>>(iv, cv, W1, b1, pi, pc);
  fused_attn<<<BSZ, 256, smemB, stream>>>(iv, cv, W2, b2, pi, pc, out);
}